// GCN_90460601188539
// MI455X (gfx1250) — compile-verified
//
#include <hip/hip_runtime.h>
#include <math.h>

// GCN 2-layer forward for MI455X (gfx1250), fp32 WMMA path.
#define NN 100000   // nodes
#define NE 800000   // edges
#define IND 128
#define HID 256
#define OD  40

typedef float v2f __attribute__((ext_vector_type(2)));
typedef float v8f __attribute__((ext_vector_type(8)));

__device__ __forceinline__ void atomAddF(float* p, float v) {
    (void)__hip_atomic_fetch_add(p, v, __ATOMIC_RELAXED, __HIP_MEMORY_SCOPE_AGENT);
}

// ---------------- degree / normalization ----------------
__global__ __launch_bounds__(256) void k_deg_init(float* __restrict__ deg) {
    int i = blockIdx.x * 256 + threadIdx.x;
    if (i < NN) deg[i] = 1.0f;   // self-loop
}

__global__ __launch_bounds__(256) void k_deg_edges(const int* __restrict__ dst,
                                                   float* __restrict__ deg) {
    int e = blockIdx.x * 256 + threadIdx.x;
    if (e < NE) atomAddF(deg + dst[e], 1.0f);
}

__global__ __launch_bounds__(256) void k_dinv(const float* __restrict__ deg,
                                              float* __restrict__ dinv) {
    int i = blockIdx.x * 256 + threadIdx.x;
    if (i < NN) dinv[i] = rsqrtf(fmaxf(deg[i], 1.0f));
}

// ---------------- GEMM1: hs1 = (X @ W1) * dinv[row] ----------------
// X:[NN,128]  W1:[128,256]  hs1:[NN,256]
// Block = 256 thr = 8 waves; wave -> 16x64 tile; block -> 32 rows x 256 cols.
// NN = 32 * 3125 exactly, no row guards needed.
__global__ __launch_bounds__(256) void k_gemm1(const float* __restrict__ X,
                                               const float* __restrict__ W,
                                               const float* __restrict__ dinv,
                                               float* __restrict__ out) {
    const int lane = threadIdx.x & 31;
    const int wave = threadIdx.x >> 5;
    const int lo   = lane & 15;   // M (for A) / N (for B,C)
    const int hi   = lane >> 4;   // K-half selector
    const int r0   = blockIdx.x * 32 + (wave >> 2) * 16;
    const int c0   = (wave & 3) * 64;

    v8f acc0 = {}, acc1 = {}, acc2 = {}, acc3 = {};

    // A: lane holds row r0+lo, K pair {k0+2*hi, k0+2*hi+1}
    const float* arow = X + (size_t)(r0 + lo) * IND + 2 * hi;

    #pragma unroll 4
    for (int k0 = 0; k0 < IND; k0 += 4) {
        __builtin_prefetch(arow + k0 + 32, 0, 0);   // global_prefetch_b8
        v2f a;
        a.x = arow[k0];
        a.y = arow[k0 + 1];
        // B: VGPR0 -> row k0+2*hi, VGPR1 -> row k0+2*hi+1; lanes carry N
        const float* bp = W + (size_t)(k0 + 2 * hi) * HID + c0 + lo;
        v2f b0, b1, b2, b3;
        b0.x = bp[0];    b0.y = bp[HID];
        b1.x = bp[16];   b1.y = bp[HID + 16];
        b2.x = bp[32];   b2.y = bp[HID + 32];
        b3.x = bp[48];   b3.y = bp[HID + 48];
        acc0 = __builtin_amdgcn_wmma_f32_16x16x4_f32(false, a, false, b0, (short)0, acc0, false, false);
        acc1 = __builtin_amdgcn_wmma_f32_16x16x4_f32(false, a, false, b1, (short)0, acc1, false, false);
        acc2 = __builtin_amdgcn_wmma_f32_16x16x4_f32(false, a, false, b2, (short)0, acc2, false, false);
        acc3 = __builtin_amdgcn_wmma_f32_16x16x4_f32(false, a, false, b3, (short)0, acc3, false, false);
    }

    // C/D layout: VGPR v -> row r0 + v + 8*hi, col c0 + lo
    #pragma unroll
    for (int v = 0; v < 8; ++v) {
        int row = r0 + v + 8 * hi;
        float s = dinv[row];
        float* orow = out + (size_t)row * HID + c0 + lo;
        orow[0]  = acc0[v] * s;
        orow[16] = acc1[v] * s;
        orow[32] = acc2[v] * s;
        orow[48] = acc3[v] * s;
    }
}

// ---------------- GEMM2: hs2 = (H @ W2) * dinv[row], N=40 (3 tiles, 3rd masked) ----
__global__ __launch_bounds__(256) void k_gemm2(const float* __restrict__ H,
                                               const float* __restrict__ W,
                                               const float* __restrict__ dinv,
                                               float* __restrict__ out) {
    const int lane = threadIdx.x & 31;
    const int wave = threadIdx.x >> 5;
    const int lo   = lane & 15;
    const int hi   = lane >> 4;
    const int r0   = blockIdx.x * 128 + wave * 16;

    const int  arow_i  = r0 + lo;
    const bool avalid  = arow_i < NN;
    const float* arow  = H + (size_t)(avalid ? arow_i : 0) * HID + 2 * hi;
    const bool cv      = lo < 8;   // third col tile covers cols 32..47, valid < 40

    v8f acc0 = {}, acc1 = {}, acc2 = {};

    #pragma unroll 4
    for (int k0 = 0; k0 < HID; k0 += 4) {
        __builtin_prefetch(arow + k0 + 32, 0, 0);
        v2f a;
        a.x = avalid ? arow[k0]     : 0.0f;
        a.y = avalid ? arow[k0 + 1] : 0.0f;
        const float* bp = W + (size_t)(k0 + 2 * hi) * OD + lo;
        v2f b0, b1, b2;
        b0.x = bp[0];               b0.y = bp[OD];
        b1.x = bp[16];              b1.y = bp[OD + 16];
        b2.x = cv ? bp[32]     : 0.0f;
        b2.y = cv ? bp[OD + 32] : 0.0f;
        acc0 = __builtin_amdgcn_wmma_f32_16x16x4_f32(false, a, false, b0, (short)0, acc0, false, false);
        acc1 = __builtin_amdgcn_wmma_f32_16x16x4_f32(false, a, false, b1, (short)0, acc1, false, false);
        acc2 = __builtin_amdgcn_wmma_f32_16x16x4_f32(false, a, false, b2, (short)0, acc2, false, false);
    }

    #pragma unroll
    for (int v = 0; v < 8; ++v) {
        int row = r0 + v + 8 * hi;
        if (row < NN) {
            float s = dinv[row];
            float* orow = out + (size_t)row * OD;
            orow[lo]      = acc0[v] * s;
            orow[16 + lo] = acc1[v] * s;
            if (cv) orow[32 + lo] = acc2[v] * s;
        }
    }
}

// ---------------- elementwise helpers ----------------
__global__ __launch_bounds__(256) void k_copy4(const float4* __restrict__ in,
                                               float4* __restrict__ out, int n4) {
    int i = blockIdx.x * 256 + threadIdx.x;
    if (i < n4) out[i] = in[i];
}

// agg1[i] = relu(agg1[i]*dinv[row] + b1[col])   (in place -> becomes H)
__global__ __launch_bounds__(256) void k_relu_ep(float* __restrict__ agg,
                                                 const float* __restrict__ dinv,
                                                 const float* __restrict__ b1) {
    int idx = blockIdx.x * 256 + threadIdx.x;     // NN*HID = 25.6M exact
    int row = idx >> 8;
    int col = idx & 255;
    float v = agg[idx] * dinv[row] + b1[col];
    agg[idx] = v > 0.0f ? v : 0.0f;
}

// ---------------- edge scatter (fp32 atomics, L2-resident accumulator) ----------
__global__ __launch_bounds__(256) void k_scatter_hid(const int* __restrict__ src,
                                                     const int* __restrict__ dst,
                                                     const float* __restrict__ hs,
                                                     float* __restrict__ agg) {
    int idx = blockIdx.x * 256 + threadIdx.x;     // NE*64 = 51.2M exact
    int e  = idx >> 6;
    int ch = idx & 63;
    int s = src[e], d = dst[e];
    float4 v = ((const float4*)(hs + (size_t)s * HID))[ch];
    float* q = agg + (size_t)d * HID + ch * 4;
    atomAddF(q + 0, v.x);
    atomAddF(q + 1, v.y);
    atomAddF(q + 2, v.z);
    atomAddF(q + 3, v.w);
}

__global__ __launch_bounds__(256) void k_scatter_out(const int* __restrict__ src,
                                                     const int* __restrict__ dst,
                                                     const float* __restrict__ hs,
                                                     float* __restrict__ agg) {
    int idx = blockIdx.x * 256 + threadIdx.x;     // NE*40 = 32M exact
    if (idx >= NE * OD) return;
    int e = idx / OD;
    int c = idx - e * OD;
    int s = src[e], d = dst[e];
    atomAddF(agg + (size_t)d * OD + c, hs[(size_t)s * OD + c]);
}

// ---------------- final: z = agg*dinv + b2 ; log_softmax, wave per row --------
__global__ __launch_bounds__(256) void k_logsoftmax(float* __restrict__ out,
                                                    const float* __restrict__ dinv,
                                                    const float* __restrict__ b2) {
    int wave = threadIdx.x >> 5;
    int lane = threadIdx.x & 31;
    int row  = blockIdx.x * 8 + wave;
    if (row >= NN) return;

    float di = dinv[row];
    float* orow = out + (size_t)row * OD;
    float z0 = orow[lane] * di + b2[lane];                       // cols 0..31
    float z1 = (lane < 8) ? orow[32 + lane] * di + b2[32 + lane] : -INFINITY;

    float m = fmaxf(z0, z1);
    #pragma unroll
    for (int o = 16; o >= 1; o >>= 1) m = fmaxf(m, __shfl_xor(m, o, 32));

    float s = expf(z0 - m) + ((lane < 8) ? expf(z1 - m) : 0.0f);
    #pragma unroll
    for (int o = 16; o >= 1; o >>= 1) s += __shfl_xor(s, o, 32);

    float lse = logf(s) + m;
    orow[lane] = z0 - lse;
    if (lane < 8) orow[32 + lane] = z1 - lse;
}

// ---------------- host ----------------
extern "C" void kernel_launch(void* const* d_in, const int* in_sizes, int n_in,
                              void* d_out, int out_size, void* d_ws, size_t ws_size,
                              hipStream_t stream) {
    const float* x   = (const float*)d_in[0];
    const int*   ei  = (const int*)d_in[1];
    const float* W1  = (const float*)d_in[2];
    const float* b1  = (const float*)d_in[3];
    const float* W2  = (const float*)d_in[4];
    const float* b2  = (const float*)d_in[5];
    float*       out = (float*)d_out;

    const int* src = ei;            // edge_index[0]
    const int* dst = ei + NE;       // edge_index[1]

    float* ws   = (float*)d_ws;
    float* dinv = ws;                                    // NN
    float* hs1  = ws + NN;                               // NN*HID
    float* agg1 = hs1 + (size_t)NN * HID;                // NN*HID (also H in-place)
    float* hs2  = agg1 + (size_t)NN * HID;               // NN*OD
    float* deg  = agg1;  // scratch: overwritten later by the self-loop copy

    // normalization
    k_deg_init <<<(NN + 255) / 256, 256, 0, stream>>>(deg);
    k_deg_edges<<<(NE + 255) / 256, 256, 0, stream>>>(dst, deg);
    k_dinv     <<<(NN + 255) / 256, 256, 0, stream>>>(deg, dinv);

    // layer 1
    k_gemm1<<<NN / 32, 256, 0, stream>>>(x, W1, dinv, hs1);
    k_copy4<<<(NN * HID / 4 + 255) / 256, 256, 0, stream>>>(
        (const float4*)hs1, (float4*)agg1, NN * HID / 4);
    k_scatter_hid<<<NE * 64 / 256, 256, 0, stream>>>(src, dst, hs1, agg1);
    k_relu_ep<<<NN * HID / 256, 256, 0, stream>>>(agg1, dinv, b1);

    // layer 2
    k_gemm2<<<(NN + 127) / 128, 256, 0, stream>>>(agg1, W2, dinv, hs2);
    k_copy4<<<(NN * OD / 4 + 255) / 256, 256, 0, stream>>>(
        (const float4*)hs2, (float4*)out, NN * OD / 4);
    k_scatter_out<<<(NE * OD + 255) / 256, 256, 0, stream>>>(src, dst, hs2, out);

    // log_softmax (in place on d_out)
    k_logsoftmax<<<(NN + 7) / 8, 256, 0, stream>>>(out, dinv, b2);
}